// SimpleLabelPropagationNoBert_83897891160442
// MI455X (gfx1250) — compile-verified
//
#include <hip/hip_runtime.h>
#include <hip/hip_bf16.h>
#include <float.h>

#define HID 256
#define EFD 20
#define ETN 3
#define NGRAPH 32
#define NCAND 100

typedef __attribute__((ext_vector_type(2))) float v2f;
typedef __attribute__((ext_vector_type(8))) float v8f;

// ---- monotone float <-> uint encoding for atomicMax on floats ----
__device__ __forceinline__ unsigned enc_f(float f) {
    unsigned b = __float_as_uint(f);
    return b ^ ((b & 0x80000000u) ? 0xFFFFFFFFu : 0x80000000u);
}
__device__ __forceinline__ float dec_f(unsigned u) {
    unsigned b = u ^ ((u & 0x80000000u) ? 0x80000000u : 0xFFFFFFFFu);
    return __uint_as_float(b);
}

// ---- init segment buffers ----
__global__ void init_kernel(unsigned* __restrict__ segmax, float* __restrict__ segsum,
                            float* __restrict__ upd, int n) {
    int i = blockIdx.x * blockDim.x + threadIdx.x;
    if (i < n) {
        segmax[i] = enc_f(-FLT_MAX);
        segsum[i] = 0.0f;
        upd[i]    = 0.0f;
    }
}

// ---- k2_type[t][i] = sum_f We[i][f] * gelu(edge_emb[t][f]) + be[i] (only ET=3 rows!) ----
__global__ __launch_bounds__(256) void k2_kernel(const float* __restrict__ We,
                                                 const float* __restrict__ be,
                                                 const float* __restrict__ edge_emb,
                                                 float* __restrict__ k2) {
    int i = threadIdx.x;  // 0..255 == HID
    for (int t = 0; t < ETN; ++t) {
        float acc = be[i];
        for (int f = 0; f < EFD; ++f) {
            float x = edge_emb[t * EFD + f];
            float g = 0.5f * x * (1.0f + erff(x * 0.70710678118654752f));  // exact GELU
            acc += We[i * EFD + f] * g;
        }
        k2[t * HID + i] = acc;
    }
}

// ---- fused dual GEMM: q = h*Wl^T + bl ; k1 = h*Wr^T + br, fp32 WMMA 16x16x4 ----
// one block = 16 nodes x 256 cols; 8 waves, 2 N-tiles each, 2 weight matrices
__global__ __launch_bounds__(256) void gemm_qk_kernel(const float* __restrict__ h,
                                                      const float* __restrict__ Wl,
                                                      const float* __restrict__ bl,
                                                      const float* __restrict__ Wr,
                                                      const float* __restrict__ br,
                                                      float* __restrict__ q,
                                                      float* __restrict__ k1) {
    __shared__ float lds[16][HID + 4];  // pad to break stride-256 bank conflicts
    const int tid  = threadIdx.x;
    const int row0 = blockIdx.x * 16;

    #pragma unroll
    for (int i = 0; i < 16; ++i)
        lds[i][tid] = h[(size_t)(row0 + i) * HID + tid];
    __syncthreads();

    const int wave  = tid >> 5;
    const int lane  = tid & 31;
    const int m     = lane & 15;
    const int khalf = (lane >> 4) << 1;      // lanes 16-31 hold K+2
    const int n0    = (wave * 2 + 0) * 16 + m;
    const int n1    = (wave * 2 + 1) * 16 + m;

    v8f accq0 = {}, accq1 = {}, acck0 = {}, acck1 = {};

    #pragma unroll 4
    for (int kk = 0; kk < 64; ++kk) {
        const int klo = kk * 4 + khalf;
        v2f a;
        a.x = lds[m][klo];
        a.y = lds[m][klo + 1];
        // B[k][n] = W[n][k]  (B = W^T), contiguous float2 along k of row n
        v2f bq0 = *(const v2f*)&Wl[(size_t)n0 * HID + klo];
        v2f bq1 = *(const v2f*)&Wl[(size_t)n1 * HID + klo];
        v2f bk0 = *(const v2f*)&Wr[(size_t)n0 * HID + klo];
        v2f bk1 = *(const v2f*)&Wr[(size_t)n1 * HID + klo];
        accq0 = __builtin_amdgcn_wmma_f32_16x16x4_f32(false, a, false, bq0, (short)0, accq0, false, false);
        accq1 = __builtin_amdgcn_wmma_f32_16x16x4_f32(false, a, false, bq1, (short)0, accq1, false, false);
        acck0 = __builtin_amdgcn_wmma_f32_16x16x4_f32(false, a, false, bk0, (short)0, acck0, false, false);
        acck1 = __builtin_amdgcn_wmma_f32_16x16x4_f32(false, a, false, bk1, (short)0, acck1, false, false);
    }

    const int mh = (lane >> 4) << 3;  // lanes 16-31 hold rows M+8
    #pragma unroll
    for (int r = 0; r < 8; ++r) {
        const size_t row = (size_t)(row0 + r + mh) * HID;
        q [row + n0] = accq0[r] + bl[n0];
        q [row + n1] = accq1[r] + bl[n1];
        k1[row + n0] = acck0[r] + br[n0];
        k1[row + n1] = acck1[r] + br[n1];
    }
}

// ---- one wave32 per edge: alpha = dot(q[dst], k1[src]+k2[etype]) / 16 ; atomic segment max ----
__global__ __launch_bounds__(256) void alpha_kernel(const float* __restrict__ q,
                                                    const float* __restrict__ k1,
                                                    const float* __restrict__ k2,
                                                    const int* __restrict__ src,
                                                    const int* __restrict__ dst,
                                                    const int* __restrict__ et,
                                                    float* __restrict__ alpha,
                                                    unsigned* __restrict__ segmax, int E) {
    const int e    = (int)((blockIdx.x * (size_t)blockDim.x + threadIdx.x) >> 5);
    const int lane = threadIdx.x & 31;
    if (e >= E) return;
    const int s = src[e], d = dst[e], t = et[e];
    const float* qp = q  + (size_t)d * HID + lane * 8;
    const float* kp = k1 + (size_t)s * HID + lane * 8;
    const float* ep = k2 + (size_t)t * HID + lane * 8;
    float v = 0.0f;
    #pragma unroll
    for (int i = 0; i < 8; ++i) v += qp[i] * (kp[i] + ep[i]);
    #pragma unroll
    for (int off = 16; off > 0; off >>= 1) v += __shfl_xor(v, off, 32);
    if (lane == 0) {
        const float al = v * 0.0625f;  // 1/sqrt(256)
        alpha[e] = al;
        atomicMax(&segmax[d], enc_f(al));
    }
}

// ---- ex = exp(alpha - max[dst]); segment sum ----
__global__ void exp_kernel(float* __restrict__ alpha, const int* __restrict__ dst,
                           const unsigned* __restrict__ segmax, float* __restrict__ segsum, int E) {
    int e = blockIdx.x * blockDim.x + threadIdx.x;
    if (e >= E) return;
    const int d = dst[e];
    const float ex = expf(alpha[e] - dec_f(segmax[d]));
    alpha[e] = ex;
    atomicAdd(&segsum[d], ex);
}

// ---- upd[dst] += scores[src] * ex / segsum[dst] ----
__global__ void agg_kernel(const float* __restrict__ alpha, const int* __restrict__ src,
                           const int* __restrict__ dst, const float* __restrict__ segsum,
                           const float* __restrict__ scores, float* __restrict__ upd, int E) {
    int e = blockIdx.x * blockDim.x + threadIdx.x;
    if (e >= E) return;
    const int d = dst[e];
    const float a = alpha[e] / segsum[d];
    atomicAdd(&upd[d], scores[src[e]] * a);
}

// ---- per-graph logits + CE ----
__global__ __launch_bounds__(128) void loss_kernel(const float* __restrict__ upd,
                                                   const int* __restrict__ cand,
                                                   const int* __restrict__ label,
                                                   float* __restrict__ out_logits,
                                                   float* __restrict__ gloss) {
    __shared__ float red[128];
    const int b = blockIdx.x, t = threadIdx.x;
    float lg = 0.0f;
    float mx = -FLT_MAX;
    if (t < NCAND) {
        lg = upd[cand[b * NCAND + t]];
        out_logits[b * NCAND + t] = lg;
        mx = lg;
    }
    red[t] = mx; __syncthreads();
    for (int s = 64; s > 0; s >>= 1) { if (t < s) red[t] = fmaxf(red[t], red[t + s]); __syncthreads(); }
    const float gmax = red[0]; __syncthreads();
    red[t] = (t < NCAND) ? expf(lg - gmax) : 0.0f; __syncthreads();
    for (int s = 64; s > 0; s >>= 1) { if (t < s) red[t] += red[t + s]; __syncthreads(); }
    if (t == 0) {
        const float lse = gmax + logf(red[0]);
        gloss[b] = lse - out_logits[b * NCAND + label[b]];  // -logp[label]
    }
}

__global__ void final_kernel(const float* __restrict__ gloss, float* __restrict__ out_loss) {
    int t = threadIdx.x;  // 32 lanes, NGRAPH==32
    float v = gloss[t];
    #pragma unroll
    for (int off = 16; off > 0; off >>= 1) v += __shfl_xor(v, off, 32);
    if (t == 0) out_loss[0] = v;
}

extern "C" void kernel_launch(void* const* d_in, const int* in_sizes, int n_in,
                              void* d_out, int out_size, void* d_ws, size_t ws_size,
                              hipStream_t stream) {
    (void)n_in; (void)out_size; (void)ws_size;
    const float* emb      = (const float*)d_in[0];
    const float* scores   = (const float*)d_in[1];
    const float* Wl       = (const float*)d_in[2];
    const float* bl       = (const float*)d_in[3];
    const float* Wr       = (const float*)d_in[4];
    const float* br       = (const float*)d_in[5];
    const float* We       = (const float*)d_in[6];
    const float* be       = (const float*)d_in[7];
    const float* edge_emb = (const float*)d_in[8];
    const int*   src      = (const int*)d_in[9];
    const int*   dst      = (const int*)d_in[10];
    const int*   etype    = (const int*)d_in[11];
    const int*   cand     = (const int*)d_in[12];
    const int*   label    = (const int*)d_in[13];

    const int N = in_sizes[1];   // scores: [N,1]
    const int E = in_sizes[9];   // src: [E]

    float* ws = (float*)d_ws;
    size_t off = 0;
    float*    attn_q = ws + off; off += (size_t)N * HID;
    float*    attn_k = ws + off; off += (size_t)N * HID;
    float*    k2     = ws + off; off += (size_t)ETN * HID;
    float*    alpha  = ws + off; off += (size_t)E;
    unsigned* segmax = (unsigned*)(ws + off); off += (size_t)N;
    float*    segsum = ws + off; off += (size_t)N;
    float*    upd    = ws + off; off += (size_t)N;
    float*    gloss  = ws + off; off += NGRAPH;

    float* out = (float*)d_out;  // [0]=loss, [1..B*C]=logits

    init_kernel<<<(N + 255) / 256, 256, 0, stream>>>(segmax, segsum, upd, N);
    k2_kernel<<<1, 256, 0, stream>>>(We, be, edge_emb, k2);
    gemm_qk_kernel<<<N / 16, 256, 0, stream>>>(emb, Wl, bl, Wr, br, attn_q, attn_k);
    alpha_kernel<<<(E + 7) / 8, 256, 0, stream>>>(attn_q, attn_k, k2, src, dst, etype, alpha, segmax, E);
    exp_kernel<<<(E + 255) / 256, 256, 0, stream>>>(alpha, dst, segmax, segsum, E);
    agg_kernel<<<(E + 255) / 256, 256, 0, stream>>>(alpha, src, dst, segsum, scores, upd, E);
    loss_kernel<<<NGRAPH, 128, 0, stream>>>(upd, cand, label, out + 1, gloss);
    final_kernel<<<1, 32, 0, stream>>>(gloss, out);
}